// FrameLabeller_43301860278431
// MI455X (gfx1250) — compile-verified
//
#include <hip/hip_runtime.h>
#include <cstring>
#include <cstdint>

#define TPB 256
#define D1V 512      // HID*HEADS
#define HIDV 128
#define NEGF (-1e30f)

typedef float v2f __attribute__((ext_vector_type(2)));
typedef float v8f __attribute__((ext_vector_type(8)));

// ---------------- utility kernels ----------------
__global__ void k_fill_f32(float* p, float v, long n) {
    long i = (long)blockIdx.x * blockDim.x + threadIdx.x;
    if (i < n) p[i] = v;
}
__global__ void k_fill_u32(unsigned* p, unsigned v, long n) {
    long i = (long)blockIdx.x * blockDim.x + threadIdx.x;
    if (i < n) p[i] = v;
}
__global__ void k_copy_u32(unsigned* d, const unsigned* s, long n) {
    long i = (long)blockIdx.x * blockDim.x + threadIdx.x;
    if (i < n) d[i] = s[i];
}

__device__ __forceinline__ unsigned fflip(float f) {
    unsigned u = __float_as_uint(f);
    return (u & 0x80000000u) ? ~u : (u | 0x80000000u);
}
__device__ __forceinline__ float funflip(unsigned u) {
    return (u & 0x80000000u) ? __uint_as_float(u & 0x7fffffffu) : __uint_as_float(~u);
}
__device__ __forceinline__ bool bit_of(const unsigned* m, int bit, int n) {
    return ((m[n] >> bit) & 1u) != 0u;
}

// ---------------- WMMA f32 GEMM with optional row gather ----------------
// C[M,Nc] = A[rowidx[m], :K] @ W[K,Nc]   (rowidx==nullptr -> identity)
// block = 256 threads = 8 waves; wave grid 2x4; block tile 32x64.
__global__ void k_gemm_wmma(const float* __restrict__ A, const int* __restrict__ rowidx,
                            const float* __restrict__ W, float* __restrict__ C,
                            int M, int K, int Nc) {
    int wave = threadIdx.x >> 5;
    int lane = threadIdx.x & 31;
    int wm = wave >> 2;          // 0..1
    int wn = wave & 3;           // 0..3
    int tileM = blockIdx.x * 32 + wm * 16;
    int tileN = blockIdx.y * 64 + wn * 16;

    int mrow = tileM + (lane & 15);
    int ka   = (lane >> 4) * 2;          // A frag K sub-offset (0 or 2)
    int ncol = tileN + (lane & 15);

    int mclamp = mrow < M ? mrow : (M - 1);
    int arow   = rowidx ? rowidx[mclamp] : mclamp;
    const float* Ap = A + (size_t)arow * K;

    v8f acc = {};
    for (int k = 0; k < K; k += 4) {
        v2f a, b;
        a.x = Ap[k + ka];
        a.y = Ap[k + ka + 1];
        b.x = W[(size_t)(k + ka) * Nc + ncol];
        b.y = W[(size_t)(k + ka + 1) * Nc + ncol];
        acc = __builtin_amdgcn_wmma_f32_16x16x4_f32(false, a, false, b,
                                                    (short)0, acc, false, false);
    }
    int rbase = tileM + (lane >> 4) * 8;
    int col   = tileN + (lane & 15);
#pragma unroll
    for (int j = 0; j < 8; ++j) {
        int r = rbase + j;
        if (r < M) C[(size_t)r * Nc + col] = acc[j];
    }
}

// ---------------- edge-attr histogram (64 bins + total in cnt[64]) ----------------
__global__ void k_hist(const int* __restrict__ eattr, const int* __restrict__ src,
                       const int* __restrict__ dst, const unsigned* mask, int bit,
                       int* cnt, int E) {
    int e = blockIdx.x * blockDim.x + threadIdx.x;
    if (e >= E) return;
    if (mask && !(bit_of(mask, bit, src[e]) && bit_of(mask, bit, dst[e]))) return;
    atomicAdd(&cnt[eattr[e]], 1);
    atomicAdd(&cnt[64], 1);
}

// meanproj[j] = sum_a cnt[a]*ap[a,j] / max(total,1)
__global__ void k_meanproj(const int* __restrict__ cnt, const float* __restrict__ ap,
                           float* __restrict__ meanp) {
    int j = blockIdx.x * blockDim.x + threadIdx.x;
    if (j >= D1V) return;
    float tot = (float)(cnt[64] > 1 ? cnt[64] : 1);
    float acc = 0.f;
    for (int a = 0; a < 64; ++a) acc += (float)cnt[a] * ap[(size_t)a * D1V + j];
    meanp[j] = acc / tot;
}

// ---------------- GATv2 raw scores: one thread per (edge, head) ----------------
__global__ void k_edge_scores(const float* __restrict__ xl, const float* __restrict__ xr,
                              const float* __restrict__ ap, const float* __restrict__ att,
                              const int* __restrict__ src, const int* __restrict__ dst,
                              const int* __restrict__ eattr, float* __restrict__ sraw, int E) {
    int t = blockIdx.x * blockDim.x + threadIdx.x;
    if (t >= E * 4) return;
    int e = t >> 2, h = t & 3;
    const float* pl = xl + (size_t)src[e] * D1V + h * HIDV;
    const float* pr = xr + (size_t)dst[e] * D1V + h * HIDV;
    const float* pe = ap + (size_t)eattr[e] * D1V + h * HIDV;
    const float* pa = att + h * HIDV;
    float acc = 0.f;
    for (int i = 0; i < HIDV; ++i) {
        float v = pl[i] + pr[i] + pe[i];
        v = v > 0.f ? v : 0.2f * v;
        acc += v * pa[i];
    }
    sraw[t] = acc;
}

// self-loop scores: one thread per (node, head); edge-emb = masked mean projection
__global__ void k_loop_scores(const float* __restrict__ xl, const float* __restrict__ xr,
                              const float* __restrict__ meanp, const float* __restrict__ att,
                              float* __restrict__ sl, int N) {
    int t = blockIdx.x * blockDim.x + threadIdx.x;
    if (t >= N * 4) return;
    int n = t >> 2, h = t & 3;
    const float* pl = xl + (size_t)n * D1V + h * HIDV;
    const float* pr = xr + (size_t)n * D1V + h * HIDV;
    const float* pe = meanp + h * HIDV;
    const float* pa = att + h * HIDV;
    float acc = 0.f;
    for (int i = 0; i < HIDV; ++i) {
        float v = pl[i] + pr[i] + pe[i];
        v = v > 0.f ? v : 0.2f * v;
        acc += v * pa[i];
    }
    sl[t] = acc;
}

// ---------------- segment softmax (max / denom) ----------------
__global__ void k_smax_edges(const float* __restrict__ sraw, const int* __restrict__ src,
                             const int* __restrict__ dst, const unsigned* mask, int bit,
                             unsigned* __restrict__ smaxU, int E) {
    int t = blockIdx.x * blockDim.x + threadIdx.x;
    if (t >= E * 4) return;
    int e = t >> 2, h = t & 3;
    if (mask && !(bit_of(mask, bit, src[e]) && bit_of(mask, bit, dst[e]))) return;
    atomicMax(&smaxU[dst[e] * 4 + h], fflip(sraw[t]));
}
__global__ void k_smax_loops(const float* __restrict__ sl, const unsigned* mask, int bit,
                             unsigned* __restrict__ smaxU, int N) {
    int t = blockIdx.x * blockDim.x + threadIdx.x;
    if (t >= N * 4) return;
    int n = t >> 2, h = t & 3;
    if (mask && !bit_of(mask, bit, n)) return;
    atomicMax(&smaxU[n * 4 + h], fflip(sl[t]));
}
__global__ void k_den_edges(const float* __restrict__ sraw, const int* __restrict__ src,
                            const int* __restrict__ dst, const unsigned* mask, int bit,
                            const unsigned* __restrict__ smaxU, float* __restrict__ den, int E) {
    int t = blockIdx.x * blockDim.x + threadIdx.x;
    if (t >= E * 4) return;
    int e = t >> 2, h = t & 3;
    if (mask && !(bit_of(mask, bit, src[e]) && bit_of(mask, bit, dst[e]))) return;
    float sm = funflip(smaxU[dst[e] * 4 + h]);
    atomicAdd(&den[dst[e] * 4 + h], expf(sraw[t] - sm));
}
__global__ void k_den_loops(const float* __restrict__ sl, const unsigned* mask, int bit,
                            const unsigned* __restrict__ smaxU, float* __restrict__ den, int N) {
    int t = blockIdx.x * blockDim.x + threadIdx.x;
    if (t >= N * 4) return;
    int n = t >> 2, h = t & 3;
    if (mask && !bit_of(mask, bit, n)) return;
    float sm = funflip(smaxU[n * 4 + h]);
    atomicAdd(&den[n * 4 + h], expf(sl[t] - sm));
}

// ---------------- aggregation ----------------
__global__ void k_outinit(float* __restrict__ out, const float* __restrict__ bias, int N) {
    long t = (long)blockIdx.x * blockDim.x + threadIdx.x;
    if (t >= (long)N * D1V) return;
    out[t] = bias[t & (D1V - 1)];
}
__global__ void k_agg_edges(const float* __restrict__ sraw, const int* __restrict__ src,
                            const int* __restrict__ dst, const unsigned* mask, int bit,
                            const unsigned* __restrict__ smaxU, const float* __restrict__ den,
                            const float* __restrict__ xl, float* __restrict__ out, int E) {
    int t = blockIdx.x * blockDim.x + threadIdx.x;
    if (t >= E * 4) return;
    int e = t >> 2, h = t & 3;
    int s = src[e], d = dst[e];
    if (mask && !(bit_of(mask, bit, s) && bit_of(mask, bit, d))) return;
    float sm = funflip(smaxU[d * 4 + h]);
    float w  = expf(sraw[t] - sm) / (den[d * 4 + h] + 1e-16f);
    const float* pl = xl + (size_t)s * D1V + h * HIDV;
    float* po = out + (size_t)d * D1V + h * HIDV;
    for (int i = 0; i < HIDV; ++i) atomicAdd(&po[i], w * pl[i]);
}
__global__ void k_agg_loops(const float* __restrict__ sl, const unsigned* mask, int bit,
                            const unsigned* __restrict__ smaxU, const float* __restrict__ den,
                            const float* __restrict__ xl, float* __restrict__ out, int N) {
    int t = blockIdx.x * blockDim.x + threadIdx.x;
    if (t >= N * 4) return;
    int n = t >> 2, h = t & 3;
    if (mask && !bit_of(mask, bit, n)) return;
    float sm = funflip(smaxU[n * 4 + h]);
    float w  = expf(sl[t] - sm) / (den[n * 4 + h] + 1e-16f);
    const float* pl = xl + (size_t)n * D1V + h * HIDV;
    float* po = out + (size_t)n * D1V + h * HIDV;
    for (int i = 0; i < HIDV; ++i) atomicAdd(&po[i], w * pl[i]);
}

// ---------------- per-channel softmax aggregation over nodes ----------------
// one block per channel; out[c] = sum_n softmax_n(masked logit)[n] * X[n,c]
__global__ void k_softmax_aggr(const float* __restrict__ X, const unsigned* mask, int bit,
                               float* __restrict__ out, int N) {
    int c = blockIdx.x;
    __shared__ float r1[TPB];
    __shared__ float r2[TPB];
    float mx = NEGF;
    for (int n = threadIdx.x; n < N; n += TPB) {
        bool m = mask ? bit_of(mask, bit, n) : true;
        float v = m ? X[(size_t)n * D1V + c] : NEGF;
        mx = fmaxf(mx, v);
    }
    r1[threadIdx.x] = mx;
    __syncthreads();
    for (int s = TPB / 2; s > 0; s >>= 1) {
        if (threadIdx.x < s) r1[threadIdx.x] = fmaxf(r1[threadIdx.x], r1[threadIdx.x + s]);
        __syncthreads();
    }
    mx = r1[0];
    __syncthreads();
    float se = 0.f, swx = 0.f;
    for (int n = threadIdx.x; n < N; n += TPB) {
        bool m = mask ? bit_of(mask, bit, n) : true;
        float x = X[(size_t)n * D1V + c];
        float v = m ? x : NEGF;
        float e = expf(v - mx);
        se += e;
        swx += e * x;
    }
    r1[threadIdx.x] = se;
    r2[threadIdx.x] = swx;
    __syncthreads();
    for (int s = TPB / 2; s > 0; s >>= 1) {
        if (threadIdx.x < s) {
            r1[threadIdx.x] += r1[threadIdx.x + s];
            r2[threadIdx.x] += r2[threadIdx.x + s];
        }
        __syncthreads();
    }
    if (threadIdx.x == 0) out[c] = r2[0] / r1[0];
}

// ---------------- k-hop: 8 masks at once as bitfield ----------------
__global__ void k_seed(const int* __restrict__ arg_nodes, unsigned* mask, int A) {
    int i = threadIdx.x;
    if (i < A) atomicOr(&mask[arg_nodes[i]], 1u << i);
}
__global__ void k_khop(const int* __restrict__ src, const int* __restrict__ dst,
                       const unsigned* __restrict__ mcur, unsigned* __restrict__ mnext, int E) {
    int e = blockIdx.x * blockDim.x + threadIdx.x;
    if (e >= E) return;
    unsigned bits = mcur[dst[e]];
    if (bits) atomicOr(&mnext[src[e]], bits);
}

// ---------------- tail: HGT + heads + log_softmax (single block) ----------------
__device__ __forceinline__ float gelu_exact(float x) {
    return 0.5f * x * (1.0f + erff(x * 0.7071067811865476f));
}
__global__ void k_tail(const float* __restrict__ framex, const float* __restrict__ rolex,
                       const float* __restrict__ kqvf, const float* __restrict__ kqvbf,
                       const float* __restrict__ kqvr, const float* __restrict__ kqvbr,
                       const float* __restrict__ a_rel, const float* __restrict__ m_rel,
                       const float* __restrict__ p_rel,
                       const float* __restrict__ aWf, const float* __restrict__ abf,
                       const float* __restrict__ aWr, const float* __restrict__ abr,
                       const float* __restrict__ Wfr, const float* __restrict__ bfr,
                       const float* __restrict__ Wro, const float* __restrict__ bro,
                       float* __restrict__ outp, int A, int NF, int NR) {
    __shared__ float kf[64], vf[64], t1[64], vfm1[64], aggf[64], gf[64], outf[64];
    __shared__ float kr[512], qr[512], vr[512], krA[512], vrM[512], aggr[512], outr[512];
    __shared__ float sfr[8], srr[8], al0[8], al1[8];
    __shared__ float red[TPB];
    __shared__ float roles_buf[512];
    const int t = threadIdx.x;
    const float sq = 0.125f;  // 1/sqrt(64)
    const float* Kf = kqvf;              const float* Vf = kqvf + 2 * 512 * 64;
    const float* Kr = kqvr;              const float* Qr = kqvr + 512 * 64;
    const float* Vr = kqvr + 2 * 512 * 64;
    const float* a1 = a_rel + 4096;      const float* a2 = a_rel + 2 * 4096;
    const float* m0 = m_rel;             const float* m1 = m_rel + 4096;
    const float* m2 = m_rel + 2 * 4096;

    // kf, vf
    if (t < 64) {
        float a = kqvbf[t];
        for (int c = 0; c < 512; ++c) a += framex[c] * Kf[(size_t)c * 64 + t];
        kf[t] = a;
    } else if (t < 128) {
        int j = t - 64;
        float a = kqvbf[128 + j];
        for (int c = 0; c < 512; ++c) a += framex[c] * Vf[(size_t)c * 64 + j];
        vf[j] = a;
    }
    __syncthreads();
    // kr, qr, vr  [A,64]
    for (int idx = t; idx < A * 64; idx += TPB) {
        int i = idx >> 6, j = idx & 63;
        float ak = kqvbr[j], aq = kqvbr[64 + j], av = kqvbr[128 + j];
        const float* rx = rolex + (size_t)i * 512;
        for (int c = 0; c < 512; ++c) {
            float x = rx[c];
            ak += x * Kr[(size_t)c * 64 + j];
            aq += x * Qr[(size_t)c * 64 + j];
            av += x * Vr[(size_t)c * 64 + j];
        }
        kr[idx] = ak; qr[idx] = aq; vr[idx] = av;
    }
    __syncthreads();
    // t1 = kf@a_rel[1], vfm1 = vf@m_rel[1], aggf = vf@m_rel[0]
    if (t < 64) {
        float a = 0.f;
        for (int k = 0; k < 64; ++k) a += kf[k] * a1[k * 64 + t];
        t1[t] = a;
    } else if (t < 128) {
        int j = t - 64; float a = 0.f;
        for (int k = 0; k < 64; ++k) a += vf[k] * m1[k * 64 + j];
        vfm1[j] = a;
    } else if (t < 192) {
        int j = t - 128; float a = 0.f;
        for (int k = 0; k < 64; ++k) a += vf[k] * m0[k * 64 + j];
        aggf[j] = a;
    }
    __syncthreads();
    // krA[i] = kr[i]@a_rel[2]; vrM[i] = vr[i]@m_rel[2]
    for (int idx = t; idx < A * 64; idx += TPB) {
        int i = idx >> 6, j = idx & 63;
        float a = 0.f, b = 0.f;
        for (int k = 0; k < 64; ++k) {
            a += kr[i * 64 + k] * a2[k * 64 + j];
            b += vr[i * 64 + k] * m2[k * 64 + j];
        }
        krA[idx] = a; vrM[idx] = b;
    }
    __syncthreads();
    if (t < A) {
        float a = 0.f, b = 0.f;
        for (int j = 0; j < 64; ++j) {
            a += qr[t * 64 + j] * t1[j];
            b += qr[t * 64 + j] * krA[t * 64 + j];
        }
        float s0 = a * sq * p_rel[1];
        float s1 = b * sq * p_rel[2];
        sfr[t] = s0; srr[t] = s1;
        float m = fmaxf(s0, s1);
        float e0 = expf(s0 - m), e1 = expf(s1 - m);
        al0[t] = e0 / (e0 + e1);
        al1[t] = e1 / (e0 + e1);
    }
    __syncthreads();
    for (int idx = t; idx < A * 64; idx += TPB) {
        int i = idx >> 6, j = idx & 63;
        aggr[idx] = al0[i] * vfm1[j] + al1[i] * vrM[idx];
    }
    if (t < 64) gf[t] = gelu_exact(aggf[t]);
    __syncthreads();
    // outf = gelu(aggf)@aWf + abf
    if (t < 64) {
        float a = abf[t];
        for (int k = 0; k < 64; ++k) a += gf[k] * aWf[k * 64 + t];
        outf[t] = a;
    }
    // gelu(aggr) in place
    for (int idx = t; idx < A * 64; idx += TPB) aggr[idx] = gelu_exact(aggr[idx]);
    __syncthreads();
    for (int idx = t; idx < A * 64; idx += TPB) {
        int i = idx >> 6, j = idx & 63;
        float a = abr[j];
        for (int k = 0; k < 64; ++k) a += aggr[i * 64 + k] * aWr[k * 64 + j];
        outr[idx] = a;
    }
    __syncthreads();
    // frame logits + log_softmax into outp[0:NF]
    float lmax = NEGF;
    for (int k = t; k < NF; k += TPB) {
        float v = bfr[k];
        for (int j = 0; j < 64; ++j) v += outf[j] * Wfr[(size_t)j * NF + k];
        outp[k] = v;
        lmax = fmaxf(lmax, v);
    }
    red[t] = lmax;
    __syncthreads();
    for (int s = TPB / 2; s > 0; s >>= 1) {
        if (t < s) red[t] = fmaxf(red[t], red[t + s]);
        __syncthreads();
    }
    float mx = red[0];
    __syncthreads();
    float lsum = 0.f;
    for (int k = t; k < NF; k += TPB) lsum += expf(outp[k] - mx);
    red[t] = lsum;
    __syncthreads();
    for (int s = TPB / 2; s > 0; s >>= 1) {
        if (t < s) red[t] += red[t + s];
        __syncthreads();
    }
    float lse = mx + logf(red[0]);
    __syncthreads();
    for (int k = t; k < NF; k += TPB) outp[k] -= lse;
    // role logits + per-row log_softmax into outp[NF : NF + A*NR]
    for (int idx = t; idx < A * NR; idx += TPB) {
        int i = idx / NR, k = idx % NR;
        float v = bro[k];
        for (int j = 0; j < 64; ++j) v += outr[i * 64 + j] * Wro[j * NR + k];
        roles_buf[idx] = v;
    }
    __syncthreads();
    if (t < A) {
        float m = NEGF;
        for (int k = 0; k < NR; ++k) m = fmaxf(m, roles_buf[t * NR + k]);
        float s = 0.f;
        for (int k = 0; k < NR; ++k) s += expf(roles_buf[t * NR + k] - m);
        float l = m + logf(s);
        for (int k = 0; k < NR; ++k) outp[NF + t * NR + k] = roles_buf[t * NR + k] - l;
    }
}

// ---------------- host side ----------------
static unsigned hflip(float f) {
    unsigned u;
    std::memcpy(&u, &f, 4);
    return (u & 0x80000000u) ? ~u : (u | 0x80000000u);
}
static inline int gdim(long n) { return (int)((n + TPB - 1) / TPB); }

extern "C" void kernel_launch(void* const* d_in, const int* in_sizes, int n_in,
                              void* d_out, int out_size, void* d_ws, size_t ws_size,
                              hipStream_t stream) {
    const int* node_pred = (const int*)d_in[0];
    const int* src       = (const int*)d_in[1];
    const int E          = in_sizes[2];
    const int* dst       = src + E;
    const int* eattr     = (const int*)d_in[2];
    const int* arg_nodes = (const int*)d_in[3];
    const float* pred_emb = (const float*)d_in[4];
    const float* Wl1 = (const float*)d_in[5];
    const float* Wr1 = (const float*)d_in[6];
    const float* We1 = (const float*)d_in[7];
    const float* att1 = (const float*)d_in[8];
    const float* b1 = (const float*)d_in[9];
    const float* Wl2 = (const float*)d_in[10];
    const float* Wr2 = (const float*)d_in[11];
    const float* We2 = (const float*)d_in[12];
    const float* att2 = (const float*)d_in[13];
    const float* b2 = (const float*)d_in[14];
    const float* kqvf = (const float*)d_in[15];
    const float* kqvbf = (const float*)d_in[16];
    const float* kqvr = (const float*)d_in[17];
    const float* kqvbr = (const float*)d_in[18];
    const float* a_rel = (const float*)d_in[19];
    const float* m_rel = (const float*)d_in[20];
    const float* p_rel = (const float*)d_in[21];
    const float* aWf = (const float*)d_in[22];
    const float* abf = (const float*)d_in[23];
    const float* aWr = (const float*)d_in[24];
    const float* abr = (const float*)d_in[25];
    const float* Wfr = (const float*)d_in[26];
    const float* bfr = (const float*)d_in[27];
    const float* Wro = (const float*)d_in[28];
    const float* bro = (const float*)d_in[29];

    const int N  = in_sizes[0];
    const int A  = in_sizes[3];
    const int NF = in_sizes[27];
    const int NR = in_sizes[29];
    const int EMB = 128;

    // workspace carve-up (float units)
    float* F = (float*)d_ws;
    size_t o = 0;
    float* BUFA = F + o; o += (size_t)N * D1V;   // xl1 -> xl2
    float* BUFB = F + o; o += (size_t)N * D1V;   // xr1 -> xr2
    float* BUFH = F + o; o += (size_t)N * D1V;   // h   -> hi (per mask)
    float* S    = F + o; o += (size_t)E * 4;     // raw edge scores
    float* SL   = F + o; o += (size_t)N * 4;     // self-loop scores
    unsigned* SMAXU = (unsigned*)(F + o); o += (size_t)N * 4;
    float* DEN  = F + o; o += (size_t)N * 4;
    float* AP1  = F + o; o += 64 * D1V;
    float* AP2  = F + o; o += 64 * D1V;
    float* MEANP = F + o; o += D1V;
    int* CNT    = (int*)(F + o); o += 80;
    unsigned* MASKA = (unsigned*)(F + o); o += (size_t)N;
    unsigned* MASKB = (unsigned*)(F + o); o += (size_t)N;
    float* FRAMEX = F + o; o += D1V;
    float* ROLEX  = F + o; o += 8 * D1V;
    if (ws_size < o * sizeof(float)) return;  // insufficient scratch

    const unsigned NEGU = hflip(NEGF);
    const dim3 blk(TPB);

    // ---- GEMMs for layer 1 (WMMA) ----
    {
        dim3 g1((64 + 31) / 32, D1V / 64);
        k_gemm_wmma<<<g1, blk, 0, stream>>>(pred_emb, nullptr, We1, AP1, 64, EMB, D1V);
        dim3 g2((N + 31) / 32, D1V / 64);
        k_gemm_wmma<<<g2, blk, 0, stream>>>(pred_emb, node_pred, Wl1, BUFA, N, EMB, D1V);
        k_gemm_wmma<<<g2, blk, 0, stream>>>(pred_emb, node_pred, Wr1, BUFB, N, EMB, D1V);
    }
    // ---- layer 1 (full mask) ----
    k_fill_u32<<<1, 128, 0, stream>>>((unsigned*)CNT, 0u, 80);
    k_hist<<<gdim(E), blk, 0, stream>>>(eattr, src, dst, nullptr, 0, CNT, E);
    k_meanproj<<<gdim(D1V), blk, 0, stream>>>(CNT, AP1, MEANP);
    k_edge_scores<<<gdim((long)E * 4), blk, 0, stream>>>(BUFA, BUFB, AP1, att1, src, dst, eattr, S, E);
    k_loop_scores<<<gdim((long)N * 4), blk, 0, stream>>>(BUFA, BUFB, MEANP, att1, SL, N);
    k_fill_u32<<<gdim((long)N * 4), blk, 0, stream>>>(SMAXU, NEGU, (long)N * 4);
    k_smax_edges<<<gdim((long)E * 4), blk, 0, stream>>>(S, src, dst, nullptr, 0, SMAXU, E);
    k_smax_loops<<<gdim((long)N * 4), blk, 0, stream>>>(SL, nullptr, 0, SMAXU, N);
    k_fill_f32<<<gdim((long)N * 4), blk, 0, stream>>>(DEN, 0.f, (long)N * 4);
    k_den_edges<<<gdim((long)E * 4), blk, 0, stream>>>(S, src, dst, nullptr, 0, SMAXU, DEN, E);
    k_den_loops<<<gdim((long)N * 4), blk, 0, stream>>>(SL, nullptr, 0, SMAXU, DEN, N);
    k_outinit<<<gdim((long)N * D1V), blk, 0, stream>>>(BUFH, b1, N);
    k_agg_edges<<<gdim((long)E * 4), blk, 0, stream>>>(S, src, dst, nullptr, 0, SMAXU, DEN, BUFA, BUFH, E);
    k_agg_loops<<<gdim((long)N * 4), blk, 0, stream>>>(SL, nullptr, 0, SMAXU, DEN, BUFA, BUFH, N);
    k_softmax_aggr<<<D1V, blk, 0, stream>>>(BUFH, nullptr, 0, FRAMEX, N);

    // ---- k-hop masks: 8 bits propagated together, 3 hops ----
    k_fill_u32<<<gdim(N), blk, 0, stream>>>(MASKA, 0u, N);
    k_seed<<<1, 32, 0, stream>>>(arg_nodes, MASKA, A);
    unsigned* mcur = MASKA;
    unsigned* mnxt = MASKB;
    for (int hop = 0; hop < 3; ++hop) {
        k_copy_u32<<<gdim(N), blk, 0, stream>>>(mnxt, mcur, N);
        k_khop<<<gdim(E), blk, 0, stream>>>(src, dst, mcur, mnxt, E);
        unsigned* tmp = mcur; mcur = mnxt; mnxt = tmp;
    }
    const unsigned* MF = mcur;

    // ---- GEMMs for layer 2 (WMMA, mask-independent; h consumed before reuse) ----
    {
        dim3 g1((64 + 31) / 32, D1V / 64);
        k_gemm_wmma<<<g1, blk, 0, stream>>>(pred_emb, nullptr, We2, AP2, 64, EMB, D1V);
        dim3 g2((N + 31) / 32, D1V / 64);
        k_gemm_wmma<<<g2, blk, 0, stream>>>(BUFH, nullptr, Wl2, BUFA, N, D1V, D1V);
        k_gemm_wmma<<<g2, blk, 0, stream>>>(BUFH, nullptr, Wr2, BUFB, N, D1V, D1V);
    }
    // raw layer-2 edge scores, computed once and reused for all masks
    k_edge_scores<<<gdim((long)E * 4), blk, 0, stream>>>(BUFA, BUFB, AP2, att2, src, dst, eattr, S, E);

    // ---- per-mask GATv2 layer 2 + masked softmax aggregation ----
    for (int i = 0; i < A; ++i) {
        k_fill_u32<<<1, 128, 0, stream>>>((unsigned*)CNT, 0u, 80);
        k_hist<<<gdim(E), blk, 0, stream>>>(eattr, src, dst, MF, i, CNT, E);
        k_meanproj<<<gdim(D1V), blk, 0, stream>>>(CNT, AP2, MEANP);
        k_loop_scores<<<gdim((long)N * 4), blk, 0, stream>>>(BUFA, BUFB, MEANP, att2, SL, N);
        k_fill_u32<<<gdim((long)N * 4), blk, 0, stream>>>(SMAXU, NEGU, (long)N * 4);
        k_smax_edges<<<gdim((long)E * 4), blk, 0, stream>>>(S, src, dst, MF, i, SMAXU, E);
        k_smax_loops<<<gdim((long)N * 4), blk, 0, stream>>>(SL, MF, i, SMAXU, N);
        k_fill_f32<<<gdim((long)N * 4), blk, 0, stream>>>(DEN, 0.f, (long)N * 4);
        k_den_edges<<<gdim((long)E * 4), blk, 0, stream>>>(S, src, dst, MF, i, SMAXU, DEN, E);
        k_den_loops<<<gdim((long)N * 4), blk, 0, stream>>>(SL, MF, i, SMAXU, DEN, N);
        k_outinit<<<gdim((long)N * D1V), blk, 0, stream>>>(BUFH, b2, N);
        k_agg_edges<<<gdim((long)E * 4), blk, 0, stream>>>(S, src, dst, MF, i, SMAXU, DEN, BUFA, BUFH, E);
        k_agg_loops<<<gdim((long)N * 4), blk, 0, stream>>>(SL, MF, i, SMAXU, DEN, BUFA, BUFH, N);
        k_softmax_aggr<<<D1V, blk, 0, stream>>>(BUFH, MF, i, ROLEX + (size_t)i * D1V, N);
    }

    // ---- tail: HGT + heads + log_softmax ----
    k_tail<<<1, TPB, 0, stream>>>(FRAMEX, ROLEX, kqvf, kqvbf, kqvr, kqvbr, a_rel, m_rel,
                                  p_rel, aWf, abf, aWr, abr, Wfr, bfr, Wro, bro,
                                  (float*)d_out, A, NF, NR);
}